// NNSARSALambda_84636625535232
// MI455X (gfx1250) — compile-verified
//
#include <hip/hip_runtime.h>
#include <hip/hip_bf16.h>
#include <math.h>

// Problem constants (from reference)
#define T_SAMP 2048
#define D_IN   128
#define H_DIM  512
#define A_DIM  8
#define DECAY  0.891f   // DISCOUNT * LAMBDA = 0.99 * 0.9

typedef __attribute__((ext_vector_type(2))) float v2f;
typedef __attribute__((ext_vector_type(8))) float v8f;

// ---------------------------------------------------------------------------
// Kernel 1: H = tanh(X @ W1 + b1)   [2048x128]@[128x512] -> [2048x512]
// One wave (32 threads) per 16x16 output tile, f32 WMMA, K-loop of 4.
// A 16x4 layout: lanes 0-15 hold K=0 (v0) / K=1 (v1); lanes 16-31 K=2/K=3.
// C/D: VGPR r = rows M=r (lanes 0-15) and M=r+8 (lanes 16-31), N = lane&15.
// ---------------------------------------------------------------------------
__global__ __launch_bounds__(32) void fwd_hidden(const float* __restrict__ X,
                                                 const float* __restrict__ W1,
                                                 const float* __restrict__ b1,
                                                 float* __restrict__ Hbuf) {
  const int m0   = blockIdx.x * 16;          // sample-tile base
  const int n0   = blockIdx.y * 16;          // hidden-tile base
  const int lane = threadIdx.x;
  const int row  = lane & 15;
  const int kk   = lane >> 4;                // 0 or 1 -> K pair select
  v8f acc = {};
#pragma unroll 4
  for (int k = 0; k < D_IN; k += 4) {
    v2f a, bm;
    const float* ap = X + (m0 + row) * D_IN + (k + 2 * kk);
    a.x  = ap[0];
    a.y  = ap[1];
    bm.x = W1[(k + 2 * kk)     * H_DIM + n0 + row];
    bm.y = W1[(k + 2 * kk + 1) * H_DIM + n0 + row];
    acc = __builtin_amdgcn_wmma_f32_16x16x4_f32(false, a, false, bm,
                                                (short)0, acc, false, false);
  }
  const int   nn   = n0 + row;
  const float bias = b1[nn];
#pragma unroll
  for (int r = 0; r < 8; ++r) {
    const int mm = m0 + r + 8 * kk;
    Hbuf[mm * H_DIM + nn] = tanhf(acc[r] + bias);
  }
}

// ---------------------------------------------------------------------------
// Kernel 2: Q = H @ W2 + b2   (A=8: scalar FMA loop; W2 is 16KB, L0-resident)
// One thread per (t, a).
// ---------------------------------------------------------------------------
__global__ void qhead(const float* __restrict__ Hbuf,
                      const float* __restrict__ W2,
                      const float* __restrict__ b2,
                      float* __restrict__ Qbuf) {
  const int tid = blockIdx.x * blockDim.x + threadIdx.x;  // [0, T*A)
  const int t = tid >> 3;
  const int a = tid & 7;
  float acc = b2[a];
  const float* hp = Hbuf + t * H_DIM;
#pragma unroll 8
  for (int h = 0; h < H_DIM; ++h) acc = fmaf(hp[h], W2[h * A_DIM + a], acc);
  Qbuf[tid] = acc;
}

// ---------------------------------------------------------------------------
// Kernel 3: errors[t] = mean_a (targets - q)^2
// ---------------------------------------------------------------------------
__global__ void td_error(const float* __restrict__ Qbuf,
                         const float* __restrict__ targets,
                         float* __restrict__ errbuf) {
  const int t = blockIdx.x * blockDim.x + threadIdx.x;
  float s = 0.0f;
#pragma unroll
  for (int a = 0; a < A_DIM; ++a) {
    const float d = targets[t * A_DIM + a] - Qbuf[t * A_DIM + a];
    s = fmaf(d, d, s);
  }
  errbuf[t] = s * (1.0f / A_DIM);
}

// ---------------------------------------------------------------------------
// Kernel 4: backward scalar scan  c_s = e_s + decay*(done_s ? 0 : c_{s+1})
// (Linear-recurrence collapse of the eligibility-trace scan.)
// ---------------------------------------------------------------------------
__global__ void back_scan(const float* __restrict__ errbuf,
                          const int* __restrict__ dones,
                          float* __restrict__ cbuf) {
  if (threadIdx.x != 0 || blockIdx.x != 0) return;
  float cnext = 0.0f;
  for (int t = T_SAMP - 1; t >= 0; --t) {
    const float cs = errbuf[t] + ((dones[t] > 0) ? 0.0f : DECAY * cnext);
    cbuf[t] = cs;
    cnext = cs;
  }
}

// ---------------------------------------------------------------------------
// Kernel 5: U[t,h] = c_t * (1 - H[t,h]^2) * W2[h, actions[t]]
// ---------------------------------------------------------------------------
__global__ void make_u(const float* __restrict__ Hbuf,
                       const float* __restrict__ W2,
                       const int* __restrict__ actions,
                       const float* __restrict__ cbuf,
                       float* __restrict__ Ubuf) {
  const int idx = blockIdx.x * blockDim.x + threadIdx.x;   // [0, T*H)
  const int t = idx >> 9;      // H_DIM == 512
  const int h = idx & 511;
  const float hv = Hbuf[idx];
  Ubuf[idx] = cbuf[t] * (1.0f - hv * hv) * W2[h * A_DIM + actions[t]];
}

// ---------------------------------------------------------------------------
// Kernel 6: grad_W1 = X^T @ U   [128x2048]@[2048x512] -> [128x512]
// f32 WMMA, K = 2048 (512 WMMAs per wave). Prefetch 64 K-steps ahead.
// ---------------------------------------------------------------------------
__global__ __launch_bounds__(32) void w1_grad(const float* __restrict__ X,
                                              const float* __restrict__ Ubuf,
                                              float* __restrict__ out) {
  const int m0   = blockIdx.x * 16;          // D tile
  const int n0   = blockIdx.y * 16;          // H tile
  const int lane = threadIdx.x;
  const int row  = lane & 15;
  const int kk   = lane >> 4;
  v8f acc = {};
#pragma unroll 2
  for (int k = 0; k < T_SAMP; k += 4) {
    if (k + 64 < T_SAMP) {
      __builtin_prefetch(X    + (k + 64 + 2 * kk) * D_IN  + m0 + row, 0, 1);
      __builtin_prefetch(Ubuf + (k + 64 + 2 * kk) * H_DIM + n0 + row, 0, 1);
    }
    v2f a, bm;
    a.x  = X[(k + 2 * kk)     * D_IN + m0 + row];      // A = X^T tile
    a.y  = X[(k + 2 * kk + 1) * D_IN + m0 + row];
    bm.x = Ubuf[(k + 2 * kk)     * H_DIM + n0 + row];
    bm.y = Ubuf[(k + 2 * kk + 1) * H_DIM + n0 + row];
    acc = __builtin_amdgcn_wmma_f32_16x16x4_f32(false, a, false, bm,
                                                (short)0, acc, false, false);
  }
#pragma unroll
  for (int r = 0; r < 8; ++r)
    out[(m0 + r + 8 * kk) * H_DIM + (n0 + row)] = acc[r];
}

// ---------------------------------------------------------------------------
// Kernel 7: grad_b1[h] = sum_t U[t,h]
// ---------------------------------------------------------------------------
__global__ void b1_grad(const float* __restrict__ Ubuf, float* __restrict__ out) {
  const int h = blockIdx.x * blockDim.x + threadIdx.x;   // [0, H)
  float s = 0.0f;
  for (int t = 0; t < T_SAMP; ++t) s += Ubuf[t * H_DIM + h];
  out[D_IN * H_DIM + h] = s;
}

// ---------------------------------------------------------------------------
// Kernel 8: grad_W2[i,j] = sum_t c_t * H[t,i] * (actions[t]==j)
// One thread per hidden unit i, 8 register accumulators (branchless select).
// ---------------------------------------------------------------------------
__global__ void w2_grad(const float* __restrict__ Hbuf,
                        const int* __restrict__ actions,
                        const float* __restrict__ cbuf,
                        float* __restrict__ out) {
  const int i = blockIdx.x * blockDim.x + threadIdx.x;   // [0, H)
  float acc[A_DIM] = {};
  for (int t = 0; t < T_SAMP; ++t) {
    const float v = cbuf[t] * Hbuf[t * H_DIM + i];
    const int a = actions[t];
#pragma unroll
    for (int j = 0; j < A_DIM; ++j) acc[j] += (a == j) ? v : 0.0f;
  }
  float* o = out + D_IN * H_DIM + H_DIM + i * A_DIM;
#pragma unroll
  for (int j = 0; j < A_DIM; ++j) o[j] = acc[j];
}

// ---------------------------------------------------------------------------
// Kernel 9: grad_b2[j] = sum_t c_t * (actions[t]==j)
// ---------------------------------------------------------------------------
__global__ void b2_grad(const int* __restrict__ actions,
                        const float* __restrict__ cbuf,
                        float* __restrict__ out) {
  const int j = threadIdx.x;
  if (j >= A_DIM) return;
  float s = 0.0f;
  for (int t = 0; t < T_SAMP; ++t) s += (actions[t] == j) ? cbuf[t] : 0.0f;
  out[D_IN * H_DIM + H_DIM + H_DIM * A_DIM + j] = s;
}

// ---------------------------------------------------------------------------
// Launch
// ---------------------------------------------------------------------------
extern "C" void kernel_launch(void* const* d_in, const int* in_sizes, int n_in,
                              void* d_out, int out_size, void* d_ws, size_t ws_size,
                              hipStream_t stream) {
  const float* xs      = (const float*)d_in[0];   // [T, D]
  const float* targets = (const float*)d_in[1];   // [T, A]
  const int*   actions = (const int*)  d_in[2];   // [T]
  const int*   dones   = (const int*)  d_in[3];   // [T]
  const float* W1      = (const float*)d_in[4];   // [D, H]
  const float* b1      = (const float*)d_in[5];   // [H]
  const float* W2      = (const float*)d_in[6];   // [H, A]
  const float* b2      = (const float*)d_in[7];   // [A]
  float* out = (float*)d_out;                     // [P = D*H + H + H*A + A]

  // Workspace layout (floats): H [T*H] | U [T*H] | Q [T*A] | err [T] | c [T]
  float* Hbuf   = (float*)d_ws;
  float* Ubuf   = Hbuf   + (size_t)T_SAMP * H_DIM;
  float* Qbuf   = Ubuf   + (size_t)T_SAMP * H_DIM;
  float* errbuf = Qbuf   + (size_t)T_SAMP * A_DIM;
  float* cbuf   = errbuf + T_SAMP;

  // 1) forward hidden GEMM (WMMA f32): (T/16) x (H/16) waves
  fwd_hidden<<<dim3(T_SAMP / 16, H_DIM / 16), 32, 0, stream>>>(xs, W1, b1, Hbuf);
  // 2) Q head
  qhead<<<(T_SAMP * A_DIM) / 256, 256, 0, stream>>>(Hbuf, W2, b2, Qbuf);
  // 3) per-sample errors
  td_error<<<T_SAMP / 256, 256, 0, stream>>>(Qbuf, targets, errbuf);
  // 4) backward trace-coefficient scan (sequential, tiny)
  back_scan<<<1, 32, 0, stream>>>(errbuf, dones, cbuf);
  // 5) U = c * (1 - H^2) * W2[:, a]
  make_u<<<(T_SAMP * H_DIM) / 256, 256, 0, stream>>>(Hbuf, W2, actions, cbuf, Ubuf);
  // 6) grad_W1 = X^T @ U (WMMA f32, K=2048)
  w1_grad<<<dim3(D_IN / 16, H_DIM / 16), 32, 0, stream>>>(xs, Ubuf, out);
  // 7) grad_b1
  b1_grad<<<H_DIM / 256, 256, 0, stream>>>(Ubuf, out);
  // 8) grad_W2
  w2_grad<<<H_DIM / 256, 256, 0, stream>>>(Hbuf, actions, cbuf, out);
  // 9) grad_b2
  b2_grad<<<1, 32, 0, stream>>>(actions, cbuf, out);
}